// RandlaNet_68539088109650
// MI455X (gfx1250) — compile-verified
//
#include <hip/hip_runtime.h>
#include <hip/hip_bf16.h>

typedef float v2f __attribute__((ext_vector_type(2)));
typedef float v8f __attribute__((ext_vector_type(8)));

#define BN_EPS 1e-6f
#define AP_WAVES 2

// ---------------------------------------------------------------------------
// Generic fused GEMM:  Y = act( bn( X@W + bias ) [+ add] )
// One wave computes one 16x16 output tile with V_WMMA_F32_16X16X4_F32.
// FLAGS: 1=bias, 2=bn, 4=act(leaky,slope), 8=residual add
// grid = ( ceil(M/16/8), ceil(Co/16) ); block = 256 (8 waves)
// Main K-loop is branch-free and software-pipelined one stage so next-step
// loads overlap the current WMMA. Tail (Ci%4) uses clamped loads + cndmask.
// ---------------------------------------------------------------------------
template <int FLAGS>
__global__ __launch_bounds__(256) void k_gemm(
    const float* __restrict__ X, const float* __restrict__ W,
    const float* __restrict__ bias,
    const float* __restrict__ gn, const float* __restrict__ bnb,
    const float* __restrict__ mn, const float* __restrict__ vn,
    const float* __restrict__ add, float* __restrict__ Y,
    int M, int Ci, int Co, float slope)
{
    const int lane = threadIdx.x & 31;
    const int wave = threadIdx.x >> 5;
    const long tm = (long)blockIdx.x * 8 + wave;
    if (tm >= (long)(M >> 4)) return;                // wave-uniform exit
    const int  tn   = blockIdx.y;
    const int  half = lane >> 4;                     // 0: K {k0,k0+1}, 1: {k0+2,k0+3}
    const int  l15  = lane & 15;
    const long mrow = tm * 16 + l15;                 // A row (M always %16==0)
    const int  col  = tn * 16 + l15;                 // B/C/D column
    const bool cok  = col < Co;
    const int  colc = cok ? col : (Co - 1);          // clamped (always valid) W col

    const float* xr = X + mrow * (long)Ci + half * 2;
    const float* w0 = W + (long)(half * 2) * Co + colc;
    const float* w1 = w0 + Co;
    const long  wstep = 4L * Co;
    const int   kmain = Ci & ~3;

    v8f acc = {0.f, 0.f, 0.f, 0.f, 0.f, 0.f, 0.f, 0.f};
    if (kmain >= 4) {                                // pipelined main loop
        v2f a = *(const v2f*)(xr);                   // global_load_b64
        v2f b;
        b.x = *w0;
        b.y = *w1;
        for (int k0 = 4; k0 < kmain; k0 += 4) {
            w0 += wstep; w1 += wstep;
            v2f a1 = *(const v2f*)(xr + k0);         // issue next-stage loads
            v2f b1;
            b1.x = *w0;
            b1.y = *w1;
            acc = __builtin_amdgcn_wmma_f32_16x16x4_f32(false, a, false, b,
                                                        (short)0, acc, false, false);
            a = a1; b = b1;
        }
        acc = __builtin_amdgcn_wmma_f32_16x16x4_f32(false, a, false, b,
                                                    (short)0, acc, false, false);
    }
    if (kmain < Ci) {                                // branch-free masked tail
        const int ka  = kmain + half * 2;
        const int ka0 = (ka     < Ci) ? ka     : (Ci - 1);
        const int ka1 = (ka + 1 < Ci) ? ka + 1 : (Ci - 1);
        const float* xb = X + mrow * (long)Ci;
        const float ax = xb[ka0];
        const float ay = xb[ka1];
        const float bx = W[(long)ka0 * Co + colc];
        const float by = W[(long)ka1 * Co + colc];
        v2f a, b;
        a.x = (ka     < Ci) ? ax : 0.f;
        a.y = (ka + 1 < Ci) ? ay : 0.f;
        b.x = (ka     < Ci) ? bx : 0.f;
        b.y = (ka + 1 < Ci) ? by : 0.f;
        acc = __builtin_amdgcn_wmma_f32_16x16x4_f32(false, a, false, b,
                                                    (short)0, acc, false, false);
    }
    if (!cok) return;

    const float pre = (FLAGS & 1) ? bias[col] : 0.f;
    float sc = 1.f, sm = 0.f, sb = 0.f;
    if (FLAGS & 2) {
        sc = gn[col] * __builtin_amdgcn_rsqf(vn[col] + BN_EPS);
        sm = mn[col]; sb = bnb[col];
    }
#pragma unroll
    for (int r = 0; r < 8; ++r) {
        const long row = tm * 16 + r + half * 8;     // C/D layout: lanes 16-31 -> M=r+8
        float val = acc[r] + pre;
        if (FLAGS & 2) val = (val - sm) * sc + sb;
        if (FLAGS & 8) val += add[row * (long)Co + col];
        if (FLAGS & 4) val = (val >= 0.f) ? val : slope * val;
        Y[row * (long)Co + col] = val;
    }
}

// ---------------------------------------------------------------------------
// Fused attention pool:
//   xcat[k,c] = concat( feat[b, idx[p,k], :Cf] , fxyz[p*16+k, :Cf] )  (C = 2*Cf)
//   y = xcat @ fcW  ((16 x C) @ (C x C)) via WMMA; att = softmax over the 16
//   neighbor rows; s[c] = sum_k xcat[k,c]*att[k,c].
// One wave per point; LDS tile stride C+1 (odd -> bank-conflict free).
// Inner K-loop software-pipelined one stage (LDS + global loads overlap WMMA).
// ---------------------------------------------------------------------------
__global__ __launch_bounds__(AP_WAVES * 32) void k_attpool(
    const float* __restrict__ feat, const int* __restrict__ idx,
    const float* __restrict__ fxyz, const float* __restrict__ fcW,
    float* __restrict__ sout, int N, long P, int Cf)
{
    const int C  = Cf * 2;
    const int SP = C + 1;
    __shared__ float lds[AP_WAVES * 16 * 257];       // supports C up to 256
    const int lane = threadIdx.x & 31;
    const int wave = threadIdx.x >> 5;
    const long p = (long)blockIdx.x * AP_WAVES + wave;
    if (p >= P) return;                              // wave-uniform exit
    float* xt = lds + wave * 16 * 257;
    const int b = (int)(p / N);
    const int* ip = idx + p * 16;

    // stage gathered + concat tile into LDS (16 x C, stride SP)
    for (int k = 0; k < 16; ++k) {
        const int j = ip[k];
        const float* fr = feat + ((long)b * N + j) * Cf;
        const float* gr = fxyz + (p * 16 + k) * (long)Cf;
        float* dst = xt + k * SP;
        for (int c = lane; c < Cf; c += 32) dst[c]      = fr[c];
        for (int c = lane; c < Cf; c += 32) dst[Cf + c] = gr[c];
    }

    const int half = lane >> 4;
    const int l15  = lane & 15;
    const int ntiles = C >> 4;
    const long wstep = 4L * C;
    const int arow = l15 * SP + half * 2;
    for (int tn = 0; tn < ntiles; ++tn) {
        const int col = tn * 16 + l15;
        const float* w0 = fcW + (long)(half * 2) * C + col;
        const float* w1 = w0 + C;
        v8f acc = {0.f, 0.f, 0.f, 0.f, 0.f, 0.f, 0.f, 0.f};
        v2f a, bf;
        a.x  = xt[arow];
        a.y  = xt[arow + 1];
        bf.x = *w0;
        bf.y = *w1;
        for (int k0 = 4; k0 < C; k0 += 4) {
            w0 += wstep; w1 += wstep;
            v2f a1, b1;
            a1.x = xt[arow + k0];
            a1.y = xt[arow + k0 + 1];
            b1.x = *w0;
            b1.y = *w1;
            acc = __builtin_amdgcn_wmma_f32_16x16x4_f32(false, a, false, bf,
                                                        (short)0, acc, false, false);
            a = a1; bf = b1;
        }
        acc = __builtin_amdgcn_wmma_f32_16x16x4_f32(false, a, false, bf,
                                                    (short)0, acc, false, false);

        // softmax over the 16 rows (neighbors); lane pair (l, l^16) share a column
        float mx = acc[0];
#pragma unroll
        for (int r = 1; r < 8; ++r) mx = fmaxf(mx, acc[r]);
        mx = fmaxf(mx, __shfl_xor(mx, 16, 32));
        float e[8];
        float ssum = 0.f;
#pragma unroll
        for (int r = 0; r < 8; ++r) { e[r] = __expf(acc[r] - mx); ssum += e[r]; }
        ssum += __shfl_xor(ssum, 16, 32);
        const float inv = __builtin_amdgcn_rcpf(ssum);   // fast v_rcp_f32
        float sp = 0.f;
#pragma unroll
        for (int r = 0; r < 8; ++r) {
            const int m = r + half * 8;
            sp += xt[m * SP + col] * (e[r] * inv);
        }
        sp += __shfl_xor(sp, 16, 32);
        if (half == 0) sout[p * (long)C + col] = sp;
    }
}

// ---------------------------------------------------------------------------
// rel_pos: out[t,0:10] = {dist, rel(3), center(3), neighbor(3)},  t over B*N*K
// ---------------------------------------------------------------------------
__global__ void k_relpos(const float* __restrict__ xyz, const int* __restrict__ idx,
                         float* __restrict__ out, int N, long R)
{
    const long t = (long)blockIdx.x * blockDim.x + threadIdx.x;
    if (t >= R) return;
    const long p = t >> 4;
    const int b = (int)(p / N);
    const int n = (int)(p % N);
    const float* c = xyz + ((long)b * N + n) * 3;
    const int j = idx[t];
    const float* nb = xyz + ((long)b * N + j) * 3;
    const float cx = c[0], cy = c[1], cz = c[2];
    const float nx = nb[0], ny = nb[1], nz = nb[2];
    const float rx = cx - nx, ry = cy - ny, rz = cz - nz;
    const float d = sqrtf(rx * rx + ry * ry + rz * rz);
    float* o = out + t * 10;
    o[0] = d;  o[1] = rx; o[2] = ry; o[3] = rz;
    o[4] = cx; o[5] = cy; o[6] = cz;
    o[7] = nx; o[8] = ny; o[9] = nz;
}

// ---------------------------------------------------------------------------
// random_sample: y[po, c] = max_k x[b, sub[po,k], c]
// ---------------------------------------------------------------------------
__global__ void k_maxpool(const float* __restrict__ x, const int* __restrict__ sub,
                          float* __restrict__ y, int Nin, int Nout, int C, long total)
{
    const long t = (long)blockIdx.x * blockDim.x + threadIdx.x;
    if (t >= total) return;
    const int  ch = (int)(t % C);
    const long po = t / C;
    const int  b  = (int)(po / Nout);
    const int* si = sub + po * 16;
    float m = -3.4e38f;
#pragma unroll
    for (int k = 0; k < 16; ++k) {
        const int j = si[k];
        m = fmaxf(m, x[((long)b * Nin + j) * C + ch]);
    }
    y[t] = m;
}

// ---------------------------------------------------------------------------
// decoder: out[pf, :Cs] = skip[pf, :] ; out[pf, Cs:] = xc[b, interp[pf], :]
// ---------------------------------------------------------------------------
__global__ void k_interp_concat(const float* __restrict__ skip, const float* __restrict__ xc,
                                const int* __restrict__ itp, float* __restrict__ out,
                                int Nf, int Nc, int Cs, int Cx, long total)
{
    const long t = (long)blockIdx.x * blockDim.x + threadIdx.x;
    if (t >= total) return;
    const int Ct = Cs + Cx;
    const int ch = (int)(t % Ct);
    const long pf = t / Ct;
    float v;
    if (ch < Cs) {
        v = skip[pf * (long)Cs + ch];
    } else {
        const int b = (int)(pf / Nf);
        const int j = itp[pf];
        v = xc[((long)b * Nc + j) * Cx + (ch - Cs)];
    }
    out[t] = v;
}

// ---------------------------------------------------------------------------
// host orchestration
// ---------------------------------------------------------------------------
struct Lin { const float* W; const float* b; };
struct BnP { const float* g; const float* b; const float* m; const float* v; };
struct ApP { const float* fcW; Lin conv; BnP bn; };
struct BbP { Lin c1; BnP b1; ApP ap1; Lin c2; BnP b2; ApP ap2; };
struct EncP { Lin c1; BnP b1; BbP bb; Lin c2; BnP b2; Lin sc; BnP bsc; };
struct DecP { Lin tc; BnP bn; };

extern "C" void kernel_launch(void* const* d_in, const int* in_sizes, int n_in,
                              void* d_out, int out_size, void* d_ws, size_t ws_size,
                              hipStream_t stream)
{
    (void)in_sizes; (void)n_in; (void)out_size; (void)ws_size;
    const int  NSv[5]  = {40960, 10240, 2560, 640, 160};
    const int  DOUT[4] = {16, 64, 128, 256};
    const int  Bc = 2;

    // ---- inputs (setup_inputs() dict order) ----
    const float* features = (const float*)d_in[0];
    const float* xyz[4]; const int* neigh[4]; const int* sub[4]; const int* itp[4];
    int ci = 1;
    for (int i = 0; i < 4; ++i) {
        xyz[i]   = (const float*)d_in[ci++];
        neigh[i] = (const int*)  d_in[ci++];
        sub[i]   = (const int*)  d_in[ci++];
        itp[i]   = (const int*)  d_in[ci++];
    }
    // ---- params: walk leaves in insertion order ----
    auto F = [&]() { return (const float*)d_in[ci++]; };
    auto getLin = [&](Lin& l) { l.W = F(); l.b = F(); };
    auto getBn  = [&](BnP& b) { b.g = F(); b.b = F(); b.m = F(); b.v = F(); };
    Lin fc1; BnP bn1; EncP enc[4]; Lin mid_c; BnP mid_bn; DecP dec[4];
    Lin c3; BnP b3; Lin c4; BnP b4; Lin c5;
    getLin(fc1); getBn(bn1);
    for (int i = 0; i < 4; ++i) {
        EncP& E = enc[i];
        getLin(E.c1); getBn(E.b1);
        getLin(E.bb.c1); getBn(E.bb.b1);
        E.bb.ap1.fcW = F(); getLin(E.bb.ap1.conv); getBn(E.bb.ap1.bn);
        getLin(E.bb.c2); getBn(E.bb.b2);
        E.bb.ap2.fcW = F(); getLin(E.bb.ap2.conv); getBn(E.bb.ap2.bn);
        getLin(E.c2); getBn(E.b2);
        getLin(E.sc); getBn(E.bsc);
    }
    getLin(mid_c); getBn(mid_bn);
    for (int i = 0; i < 4; ++i) { getLin(dec[i].tc); getBn(dec[i].bn); }
    getLin(c3); getBn(b3); getLin(c4); getBn(b4); getLin(c5);

    // ---- workspace bump allocator ----
    char* wsp = (char*)d_ws; size_t off = 0;
    auto alloc = [&](long floats) -> float* {
        float* p = (float*)(wsp + off);
        off += (size_t)((floats * 4 + 255) / 256) * 256;
        return p;
    };
    float* xA   = alloc(2621440);                      // up to 81920 x 32
    float* xB   = alloc(5242880);                      // up to 81920 x 64 / decoder concat
    float* enc0 = alloc((long)Bc * NSv[0] * 32);       // DRB0 output (skip)
    float* encS[4];
    encS[0] = alloc((long)Bc * NSv[1] * 32);
    encS[1] = alloc((long)Bc * NSv[2] * 128);
    encS[2] = alloc((long)Bc * NSv[3] * 256);
    encS[3] = alloc((long)Bc * NSv[4] * 512);
    float* fbuf  = alloc(655360);                      // P x h
    float* relp  = alloc(13107200);                    // R x 10
    float* fx1   = alloc(10485760);                    // R x h
    float* fx2   = alloc(10485760);                    // R x h
    float* s1    = alloc(1310720);                     // P x 2h
    float* aggb  = alloc(655360);                      // P x h
    float* s2    = alloc(1310720);                     // P x 2h
    float* bbout = alloc(1310720);                     // P x d
    float* f2b   = alloc(2621440);                     // P x 2d

    auto gemm = [&](const float* X, float* Y, long M, int Ci_, int Co,
                    const Lin& lin, const BnP* bn, bool act, float slope,
                    const float* add) {
        int flags = 1;
        if (bn)  flags |= 2;
        if (act) flags |= 4;
        if (add) flags |= 8;
        dim3 grid((unsigned)((M / 16 + 7) / 8), (unsigned)((Co + 15) / 16));
        const float* g_ = bn ? bn->g : nullptr;
        const float* bb_ = bn ? bn->b : nullptr;
        const float* m_ = bn ? bn->m : nullptr;
        const float* v_ = bn ? bn->v : nullptr;
        switch (flags) {
        case 1:
            k_gemm<1><<<grid, 256, 0, stream>>>(X, lin.W, lin.b, g_, bb_, m_, v_,
                                                add, Y, (int)M, Ci_, Co, slope);
            break;
        case 3:
            k_gemm<3><<<grid, 256, 0, stream>>>(X, lin.W, lin.b, g_, bb_, m_, v_,
                                                add, Y, (int)M, Ci_, Co, slope);
            break;
        case 7:
            k_gemm<7><<<grid, 256, 0, stream>>>(X, lin.W, lin.b, g_, bb_, m_, v_,
                                                add, Y, (int)M, Ci_, Co, slope);
            break;
        default:
            k_gemm<15><<<grid, 256, 0, stream>>>(X, lin.W, lin.b, g_, bb_, m_, v_,
                                                 add, Y, (int)M, Ci_, Co, slope);
            break;
        }
    };
    auto attpool = [&](const float* feat, const int* nidx, const float* fxv,
                       const float* fcW, float* sout, int N, long P, int Cf) {
        const int blocks = (int)((P + AP_WAVES - 1) / AP_WAVES);
        k_attpool<<<blocks, AP_WAVES * 32, 0, stream>>>(feat, nidx, fxv, fcW,
                                                        sout, N, P, Cf);
    };

    // ---- fc1 + bn1 + lrelu ----
    gemm(features, xA, (long)Bc * NSv[0], 6, 8, fc1, &bn1, true, 0.2f, nullptr);
    const float* x = xA;
    int f_in = 8;

    // ---- encoder ----
    for (int i = 0; i < 4; ++i) {
        const int N = NSv[i];
        const long Pp = (long)Bc * N;
        const long R  = Pp * 16;
        const int h = DOUT[i] / 2, d = DOUT[i], dd = 2 * d;
        EncP& E = enc[i];
        gemm(x, fbuf, Pp, f_in, h, E.c1, &E.b1, true, 0.2f, nullptr);
        {
            const int blocks = (int)((R + 255) / 256);
            k_relpos<<<blocks, 256, 0, stream>>>(xyz[i], neigh[i], relp, N, R);
        }
        gemm(relp, fx1, R, 10, h, E.bb.c1, &E.bb.b1, true, 0.2f, nullptr);
        attpool(fbuf, neigh[i], fx1, E.bb.ap1.fcW, s1, N, Pp, h);
        gemm(s1, aggb, Pp, 2 * h, h, E.bb.ap1.conv, &E.bb.ap1.bn, true, 0.2f, nullptr);
        gemm(fx1, fx2, R, h, h, E.bb.c2, &E.bb.b2, true, 0.2f, nullptr);
        attpool(aggb, neigh[i], fx2, E.bb.ap2.fcW, s2, N, Pp, h);
        gemm(s2, bbout, Pp, 2 * h, d, E.bb.ap2.conv, &E.bb.ap2.bn, true, 0.2f, nullptr);
        gemm(bbout, f2b, Pp, d, dd, E.c2, &E.b2, false, 0.f, nullptr);
        float* drbout = (i == 0) ? enc0 : xA;
        gemm(x, drbout, Pp, f_in, dd, E.sc, &E.bsc, true, 0.01f, f2b);
        {
            const long Pn = (long)Bc * NSv[i + 1];
            const long total = Pn * dd;
            const int blocks = (int)((total + 255) / 256);
            k_maxpool<<<blocks, 256, 0, stream>>>(drbout, sub[i], encS[i],
                                                  N, NSv[i + 1], dd, total);
        }
        x = encS[i];
        f_in = dd;
    }

    // ---- mid ----
    gemm(x, xA, (long)Bc * NSv[4], 512, 512, mid_c, &mid_bn, true, 0.2f, nullptr);
    x = xA;

    // ---- decoder ----
    const float* skips[4] = { encS[2], encS[1], encS[0], enc0 };
    const int Ns_f[4] = { NSv[3], NSv[2], NSv[1], NSv[0] };
    const int Ns_c[4] = { NSv[4], NSv[3], NSv[2], NSv[1] };
    const int Cs_[4]  = { 256, 128, 32, 32 };
    const int Cx_[4]  = { 512, 256, 128, 32 };
    const int Fo_[4]  = { 256, 128, 32, 32 };
    for (int i = 0; i < 4; ++i) {
        const long Pf = (long)Bc * Ns_f[i];
        const long total = Pf * (Cs_[i] + Cx_[i]);
        const int blocks = (int)((total + 255) / 256);
        k_interp_concat<<<blocks, 256, 0, stream>>>(skips[i], x, itp[3 - i], xB,
                                                    Ns_f[i], Ns_c[i], Cs_[i], Cx_[i], total);
        gemm(xB, xA, Pf, Cs_[i] + Cx_[i], Fo_[i], dec[i].tc, &dec[i].bn,
             false, 0.f, nullptr);
        x = xA;
    }

    // ---- heads ----
    const long P0 = (long)Bc * NSv[0];
    gemm(x, xB, P0, 32, 64, c3, &b3, true, 0.2f, nullptr);
    gemm(xB, xA, P0, 64, 32, c4, &b4, true, 0.2f, nullptr);
    gemm(xA, (float*)d_out, P0, 32, 13, c5, nullptr, false, 0.f, nullptr);
}